// mha_SMC_10471130268289
// MI455X (gfx1250) — compile-verified
//
#include <hip/hip_runtime.h>
#include <hip/hip_bf16.h>
#include <math.h>

// Problem constants (compile-time, from the reference)
#define B_      4
#define P_      8
#define S_      2048
#define D_      1024
#define H_      16
#define DEPTH_  64
#define WINDOW_ 64
#define BP_     (B_ * P_)          // 32 decode tokens
#define CACHE_N (BP_ * S_ * D_)    // 67,108,864 floats per cache tensor

typedef float v2f __attribute__((ext_vector_type(2)));
typedef float v4f __attribute__((ext_vector_type(4)));   // native vector: OK for NT builtins
typedef float v8f __attribute__((ext_vector_type(8)));

// ---------------------------------------------------------------------------
// GEMM: C[32 x 1024] = X[32 x 1024] @ W[1024 x 1024] + bias, fp32 via
// V_WMMA_F32_16X16X4_F32. One wave (32 lanes) owns one 16x16 C tile.
// A-frag (16x4 f32): VGPR j, lane-half h -> K = k0 + 2h + j, M = lane&15.
// B-frag (4x16 f32): mirrored (N = lane&15).
// C/D: VGPR j -> M = j + 8*half, N = lane&15.
// ---------------------------------------------------------------------------
__global__ __launch_bounds__(32)
void gemm_wmma_f32(const float* __restrict__ X,
                   const float* __restrict__ W,
                   const float* __restrict__ bias,
                   float* __restrict__ C)
{
    const int lane = threadIdx.x & 31;
    const int half = lane >> 4;        // 0 or 1
    const int lm   = lane & 15;
    const int n0   = blockIdx.x * 16;  // gridDim.x = 64
    const int m0   = blockIdx.y * 16;  // gridDim.y = 2

    const float* xrow = X + (m0 + lm) * D_;   // A row for this lane
    v8f acc = {};

#pragma unroll 4
    for (int k0 = 0; k0 < D_; k0 += 4) {
        const int ka = k0 + 2 * half;
        v2f a, b;
        a.x = xrow[ka + 0];
        a.y = xrow[ka + 1];
        b.x = W[(ka + 0) * D_ + n0 + lm];
        b.y = W[(ka + 1) * D_ + n0 + lm];
        acc = __builtin_amdgcn_wmma_f32_16x16x4_f32(
            /*neg_a=*/false, a, /*neg_b=*/false, b,
            /*c_mod=*/(short)0, acc, /*reuse_a=*/false, /*reuse_b=*/false);
    }

    const float bb = bias[n0 + lm];
#pragma unroll
    for (int j = 0; j < 8; ++j) {
        const int m = m0 + j + 8 * half;
        C[m * D_ + n0 + lm] = acc[j] + bb;
    }
}

// ---------------------------------------------------------------------------
// Streaming cache copy: K->outK, V->outV. This is the bandwidth-dominant part
// (~1 GB total). B128 non-temporal loads/stores to bypass cache retention.
// blockIdx.z selects K (0) or V (1).
// ---------------------------------------------------------------------------
__global__ __launch_bounds__(256)
void cache_copy(const v4f* __restrict__ srcK, const v4f* __restrict__ srcV,
                v4f* __restrict__ dstK, v4f* __restrict__ dstV)
{
    const v4f* __restrict__ src = (blockIdx.z == 0) ? srcK : srcV;
    v4f* __restrict__ dst       = (blockIdx.z == 0) ? dstK : dstV;

    const unsigned total4 = (unsigned)(CACHE_N / 4);   // 16,777,216
    const unsigned stride = gridDim.x * blockDim.x;
    for (unsigned i = blockIdx.x * blockDim.x + threadIdx.x; i < total4; i += stride) {
        v4f v = __builtin_nontemporal_load(&src[i]);
        __builtin_nontemporal_store(v, &dst[i]);
    }
}

// Overwrite cache row `t` for each (b,p) with the freshly projected k/v.
// 32 blocks (one per bp) x 256 threads; 256 x v4f = one 1024-float row.
__global__ __launch_bounds__(256)
void write_new_rows(const float* __restrict__ kt, const float* __restrict__ vt,
                    float* __restrict__ outK, float* __restrict__ outV,
                    const int* __restrict__ tptr)
{
    const int t  = *tptr;
    const int bp = blockIdx.x;
    const size_t rowoff = ((size_t)bp * S_ + (size_t)t) * D_;

    const v4f* k4 = (const v4f*)(kt + bp * D_);
    const v4f* v4 = (const v4f*)(vt + bp * D_);
    v4f* ok = (v4f*)(outK + rowoff);
    v4f* ov = (v4f*)(outV + rowoff);
    ok[threadIdx.x] = k4[threadIdx.x];
    ov[threadIdx.x] = v4[threadIdx.x];
}

// ---------------------------------------------------------------------------
// Sliding-window attention for the single decode token. One wave32 per head.
// Reads the UPDATED caches (outK/outV) so row t already holds the new k/v.
// Writes the full attn row (exact zeros outside window == fp32 softmax of
// -1e9-masked logits) and the per-head context into Zh.
// ---------------------------------------------------------------------------
__global__ __launch_bounds__(32)
void attention_window(const float* __restrict__ Q,    // (32, 1024)
                      const float* __restrict__ Kc,   // (32, S, 1024) updated
                      const float* __restrict__ Vc,   // (32, S, 1024) updated
                      float* __restrict__ attn,       // (32, H, S)
                      float* __restrict__ Zh,         // (32, 1024)
                      const int* __restrict__ tptr)
{
    __shared__ float smQ[DEPTH_];
    __shared__ float smP[96];

    const int lane = threadIdx.x;          // 0..31
    const int head = blockIdx.x % H_;
    const int bp   = blockIdx.x / H_;
    const int t    = *tptr;
    const int lo   = (t > WINDOW_) ? (t - WINDOW_) : 0;
    const int cnt  = t - lo + 1;           // <= 65

    // stage q for this head in LDS
    const float* q = Q + bp * D_ + head * DEPTH_;
    smQ[lane]      = q[lane];
    smQ[lane + 32] = q[lane + 32];
    __syncthreads();

    const float scale = 0.125f;            // 1/sqrt(DEPTH)

    // each lane owns window positions lo + lane + 32*j, j = 0..2
    float lg[3];
    float mymax = -INFINITY;
#pragma unroll
    for (int j = 0; j < 3; ++j) {
        const int rel = lane + 32 * j;
        float acc = -INFINITY;
        if (rel < cnt) {
            const float* krow =
                Kc + ((size_t)bp * S_ + (size_t)(lo + rel)) * D_ + head * DEPTH_;
            float s = 0.f;
#pragma unroll
            for (int d = 0; d < DEPTH_; ++d) s += smQ[d] * krow[d];
            acc = s * scale;
        }
        lg[j] = acc;
        mymax = fmaxf(mymax, acc);
    }
    // wave32 max reduction
    for (int off = 16; off > 0; off >>= 1)
        mymax = fmaxf(mymax, __shfl_xor(mymax, off, 32));

    float pv[3];
    float mysum = 0.f;
#pragma unroll
    for (int j = 0; j < 3; ++j) {
        const float p = (lg[j] == -INFINITY) ? 0.f : __expf(lg[j] - mymax);
        pv[j] = p;
        mysum += p;
    }
    for (int off = 16; off > 0; off >>= 1)
        mysum += __shfl_xor(mysum, off, 32);
    const float inv = 1.f / mysum;

#pragma unroll
    for (int j = 0; j < 3; ++j)
        smP[lane + 32 * j] = pv[j] * inv;
    __syncthreads();

    // write the full attn row: zeros outside [lo, t]
    float* arow = attn + ((size_t)(bp * H_ + head)) * S_;
    for (int s = lane; s < S_; s += 32) {
        const int rel = s - lo;
        arow[s] = (rel >= 0 && rel < cnt) ? smP[rel] : 0.f;
    }

    // z[d] = sum_s p[s] * V[s][d]; lane owns dims {lane, lane+32}
    float z0 = 0.f, z1 = 0.f;
    for (int r = 0; r < cnt; ++r) {
        const float* vrow =
            Vc + ((size_t)bp * S_ + (size_t)(lo + r)) * D_ + head * DEPTH_;
        const float p = smP[r];
        z0 += p * vrow[lane];
        z1 += p * vrow[lane + 32];
    }
    float* zrow = Zh + bp * D_ + head * DEPTH_;
    zrow[lane]      = z0;
    zrow[lane + 32] = z1;
}

// ---------------------------------------------------------------------------
extern "C" void kernel_launch(void* const* d_in, const int* in_sizes, int n_in,
                              void* d_out, int out_size, void* d_ws, size_t ws_size,
                              hipStream_t stream)
{
    const float* x   = (const float*)d_in[0];   // (B,P,1,D) = (32,1024)
    const float* K   = (const float*)d_in[1];   // (32,S,1024)
    const float* V   = (const float*)d_in[2];
    const float* Wq  = (const float*)d_in[3];
    const float* bq  = (const float*)d_in[4];
    const float* Wk  = (const float*)d_in[5];
    const float* bk  = (const float*)d_in[6];
    const float* Wv  = (const float*)d_in[7];
    const float* bv  = (const float*)d_in[8];
    const float* Wo  = (const float*)d_in[9];
    const float* bo  = (const float*)d_in[10];
    const int*  tptr = (const int*)d_in[11];

    float* out  = (float*)d_out;
    float* z    = out;                              // 32768
    float* outK = z + (size_t)BP_ * D_;             // 67,108,864
    float* outV = outK + (size_t)CACHE_N;           // 67,108,864
    float* attn = outV + (size_t)CACHE_N;           // 1,048,576

    float* ws  = (float*)d_ws;
    float* Qb  = ws;                                //  32x1024
    float* Ktb = Qb  + (size_t)BP_ * D_;
    float* Vtb = Ktb + (size_t)BP_ * D_;
    float* Zh  = Vtb + (size_t)BP_ * D_;

    const dim3 gGemm(D_ / 16, BP_ / 16, 1);         // (64, 2)

    // 1) QKV projections (WMMA fp32)
    gemm_wmma_f32<<<gGemm, 32, 0, stream>>>(x, Wq, bq, Qb);
    gemm_wmma_f32<<<gGemm, 32, 0, stream>>>(x, Wk, bk, Ktb);
    gemm_wmma_f32<<<gGemm, 32, 0, stream>>>(x, Wv, bv, Vtb);

    // 2) bandwidth-dominant cache copy (NT B128), then patch row t
    cache_copy<<<dim3(4096, 1, 2), 256, 0, stream>>>(
        (const v4f*)K, (const v4f*)V, (v4f*)outK, (v4f*)outV);
    write_new_rows<<<BP_, 256, 0, stream>>>(Ktb, Vtb, outK, outV, tptr);

    // 3) sliding-window attention, one wave per head
    attention_window<<<BP_ * H_, 32, 0, stream>>>(Qb, outK, outV, attn, Zh, tptr);

    // 4) output projection (WMMA fp32)
    gemm_wmma_f32<<<gGemm, 32, 0, stream>>>(Zh, Wo, bo, z);
}